// MPLinear_53506702573939
// MI455X (gfx1250) — compile-verified
//
#include <hip/hip_runtime.h>

#define NB    16      // batch
#define H1    1024    // inner dim (D)
#define H2    1024    // output dim
#define GAMMA_F 1.0f

// Full-wave32 butterfly sum; result broadcast to all lanes.
__device__ __forceinline__ float wave_reduce_add(float v) {
#pragma unroll
  for (int m = 16; m >= 1; m >>= 1) v += __shfl_xor(v, m, 32);
  return v;
}

// Michelot's exact sparsemax-threshold solve over the 1024 values held as
// 32 floats/lane across the wave: find tau s.t. sum relu(a_d - tau) = gamma.
// (Equivalent to the reference's sort/cumsum/mask pipeline; tau > 0 so the
// mirrored negative half of the 2048-set never enters the support.)
__device__ __forceinline__ float sparsemax_threshold(const float (&a)[32], float gamma) {
  float S = 0.f;
#pragma unroll
  for (int j = 0; j < 32; ++j) S += a[j];
  S = wave_reduce_add(S);
  float tau   = (S - gamma) * (1.0f / (float)H1);
  float kprev = (float)H1;
#pragma unroll 1
  for (int it = 0; it < 32; ++it) {      // converges in ~5-8 iters for this data
    float s = 0.f, k = 0.f;
#pragma unroll
    for (int j = 0; j < 32; ++j) {
      const bool act = a[j] > tau;
      s += act ? a[j] : 0.f;
      k += act ? 1.f : 0.f;
    }
    s = wave_reduce_add(s);
    k = wave_reduce_add(k);              // support never empties: max stays active
    tau = (s - gamma) / k;
    if (k == kprev) break;               // active set unchanged -> fixed point
    kprev = k;
  }
  return tau;
}

// One block per output column n; 16 waves, wave b handles batch row b.
extern "C" __global__ __launch_bounds__(512)
void mp_linear_sparsemax_kernel(const float* __restrict__ x,      // (16,1024)
                                const float* __restrict__ weight, // (1024,1024)
                                float* __restrict__ out) {        // (16,1024)
  __shared__ float wrow[H1];             // 4KB: weight row n, shared by 16 waves
  const int n    = blockIdx.x;
  const int tid  = threadIdx.x;
  const int lane = tid & 31;
  const int b    = tid >> 5;             // wave id == batch row

#if defined(__gfx1250__)
  // CDNA5 async copy: waves 0..7 stage the 4KB weight row into LDS
  // (16B per lane * 256 lanes), tracked by ASYNCcnt.
  if (tid < 256) {
    const float* gsrc = weight + (size_t)n * H1 + (size_t)tid * 4;
    typedef __attribute__((address_space(3))) float lds_float_t;
    lds_float_t* ldst = (lds_float_t*)(&wrow[tid * 4]);
    asm volatile("global_load_async_to_lds_b128 %0, %1, off"
                 :: "v"(ldst), "v"(gsrc) : "memory");
  }
  // Warm L2/L0 for this wave's x row while the async copy is in flight.
  __builtin_prefetch(x + b * H1 + lane * 32, 0, 1);   // global_prefetch_b8
  asm volatile("s_wait_asynccnt 0x0" ::: "memory");   // issuers drain ASYNCcnt
#else
  for (int i = tid; i < H1; i += 512) wrow[i] = weight[(size_t)n * H1 + i];
#endif
  __syncthreads();

  // Build |w+x| and |w-x|: 32 values per lane -> whole 1024-row per wave.
  float a[32], c[32];
  const float* xb = x + b * H1;
#pragma unroll
  for (int j = 0; j < 32; ++j) {
    const int d   = j * 32 + lane;       // stride-1 across lanes: coalesced / bank-free
    const float wv = wrow[d];
    const float xv = xb[d];
    a[j] = fabsf(wv + xv);
    c[j] = fabsf(wv - xv);
  }

  const float alpha = sparsemax_threshold(a, GAMMA_F);
  const float beta  = sparsemax_threshold(c, GAMMA_F);

  if (lane == 0)
    out[b * H2 + n] = 0.5f * GAMMA_F * (float)H1 * (alpha - beta);
}

extern "C" void kernel_launch(void* const* d_in, const int* in_sizes, int n_in,
                              void* d_out, int out_size, void* d_ws, size_t ws_size,
                              hipStream_t stream) {
  const float* x      = (const float*)d_in[0];   // (16,1024) f32
  const float* weight = (const float*)d_in[1];   // (1024,1024) f32
  float* out          = (float*)d_out;           // (16,1024) f32
  (void)in_sizes; (void)n_in; (void)out_size; (void)d_ws; (void)ws_size;
  dim3 grid(H2), block(512);
  hipLaunchKernelGGL(mp_linear_sparsemax_kernel, grid, block, 0, stream, x, weight, out);
}